// SUBJ_DKVMN_76828374991741
// MI455X (gfx1250) — compile-verified
//
#include <hip/hip_runtime.h>
#include <hip/hip_bf16.h>
#include <math.h>

#define BB 128
#define NN 200
#define DD 64
#define MM 50
#define TT (BB * NN)   // 25600 tokens

typedef float v2f __attribute__((ext_vector_type(2)));
typedef float v8f __attribute__((ext_vector_type(8)));

// ---------------------------------------------------------------------------
// 1) Gather K = k_emb[q], V = v_emb[q + r*1000]  (float4 vectorized)
// ---------------------------------------------------------------------------
__global__ void dkvmn_gather(const int* __restrict__ q, const int* __restrict__ r,
                             const float* __restrict__ k_emb, const float* __restrict__ v_emb,
                             float* __restrict__ Kg, float* __restrict__ Vg) {
    int gid = blockIdx.x * 256 + threadIdx.x;     // one float4 per thread
    if (gid >= TT * (DD / 4)) return;
    int bt = gid >> 4;            // token
    int d4 = gid & 15;            // float4 index within the 64-dim row
    int qi = q[bt];
    int ri = r[bt];
    const float4* ke = (const float4*)(k_emb) + qi * (DD / 4) + d4;
    const float4* ve = (const float4*)(v_emb) + (qi + ri * 1000) * (DD / 4) + d4;
    ((float4*)Kg)[gid] = *ke;
    ((float4*)Vg)[gid] = *ve;
}

// ---------------------------------------------------------------------------
// 2) f32 WMMA GEMM (compile-time shape): out[TT,N] = act(X[TT,64] @ B + bias)
//    B[k][n] = TR ? W[n*64 + k] (W is [N,64]) : W[k*N + n] (W is [64,N])
//    ACT: 0 = none, 1 = sigmoid, 2 = tanh.  One wave per 16-row tile.
// ---------------------------------------------------------------------------
template <int N, bool TR, int ACT>
__global__ void dkvmn_gemm_act(const float* __restrict__ X, const float* __restrict__ W,
                               const float* __restrict__ bias, float* __restrict__ out) {
    constexpr int NT = (N + 15) / 16;
    int wave = threadIdx.x >> 5;
    int lane = threadIdx.x & 31;
    int half = lane >> 4;           // 0: lanes 0-15, 1: lanes 16-31
    int l    = lane & 15;
    int row0 = (blockIdx.x * 4 + wave) * 16;
    if (row0 >= TT) return;         // wave-uniform

    v8f acc[NT] = {};
    const float* Xr = X + (size_t)(row0 + l) * DD;

#pragma unroll
    for (int k0 = 0; k0 < DD; k0 += 4) {
        int k = k0 + 2 * half;
        v2f a;
        a.x = Xr[k + 0];
        a.y = Xr[k + 1];
#pragma unroll
        for (int nt = 0; nt < NT; ++nt) {
            int n = nt * 16 + l;
            v2f b;
            if (TR) {
                int ncl = (n < N) ? n : 0;
                float b0 = W[(size_t)ncl * DD + k + 0];
                float b1 = W[(size_t)ncl * DD + k + 1];
                b.x = (n < N) ? b0 : 0.0f;
                b.y = (n < N) ? b1 : 0.0f;
            } else {
                b.x = W[(size_t)(k + 0) * N + n];
                b.y = W[(size_t)(k + 1) * N + n];
            }
            acc[nt] = __builtin_amdgcn_wmma_f32_16x16x4_f32(
                false, a, false, b, (short)0, acc[nt], false, false);
        }
    }

#pragma unroll
    for (int nt = 0; nt < NT; ++nt) {
        int n = nt * 16 + l;
        if (n < N) {
            float bs = bias ? bias[n] : 0.0f;
#pragma unroll
            for (int i = 0; i < 8; ++i) {
                int m = i + 8 * half;
                float val = acc[nt][i] + bs;
                if (ACT == 1)      val = 1.0f / (1.0f + __expf(-val));
                else if (ACT == 2) val = tanhf(val);
                out[(size_t)(row0 + m) * N + n] = val;
            }
        }
    }
}

// ---------------------------------------------------------------------------
// 3) Row softmax over 50 logits, in place
// ---------------------------------------------------------------------------
__global__ void dkvmn_softmax(float* __restrict__ Wb) {
    int t = blockIdx.x * 256 + threadIdx.x;
    if (t >= TT) return;
    float* row = Wb + (size_t)t * MM;
    float mx = row[0];
    for (int m = 1; m < MM; ++m) mx = fmaxf(mx, row[m]);
    float s = 0.0f;
    for (int m = 0; m < MM; ++m) s += __expf(row[m] - mx);
    float inv = 1.0f / s;
    for (int m = 0; m < MM; ++m) row[m] = __expf(row[m] - mx) * inv;
}

// ---------------------------------------------------------------------------
// 4) Sequential scan. One block (256 threads) per batch element.
//    Mv state (50x64 f32 = 12.8KB) in LDS. Per step:
//      a) stream pre-update Mv to MvOut[b][t] as float4 (dominant HBM traffic)
//      b) read[d] = sum_m w[m] * Mv[m][d]        (pre-update)
//      c) Mv[m][d] = Mv[m][d]*(1 - w[m]e[d]) + w[m]a[d]
// ---------------------------------------------------------------------------
__global__ void dkvmn_scan(const float* __restrict__ Mv0, const float* __restrict__ Wg,
                           const float* __restrict__ Eg, const float* __restrict__ Ag,
                           float* __restrict__ Rd, float* __restrict__ MvOut) {
    __shared__ __align__(16) float mv[MM * DD];
    __shared__ float wsh[MM];
    __shared__ float esh[DD];
    __shared__ float ash[DD];

    int b   = blockIdx.x;
    int tid = threadIdx.x;
    const float4* mv4 = (const float4*)mv;

    // init state from Mv0 (float4)
    for (int i = tid; i < MM * DD / 4; i += 256)
        ((float4*)mv)[i] = ((const float4*)Mv0)[i];
    __syncthreads();

    for (int t = 0; t < NN; ++t) {
        int bt = b * NN + t;
        // load gates for this step
        if (tid < MM)                 wsh[tid]           = Wg[(size_t)bt * MM + tid];
        else if (tid < MM + DD)       esh[tid - MM]      = Eg[(size_t)bt * DD + (tid - MM)];
        else if (tid < MM + 2 * DD)   ash[tid - MM - DD] = Ag[(size_t)bt * DD + (tid - MM - DD)];
        // stream pre-update state to output (b128 stores)
        float4* dst4 = (float4*)(MvOut + ((size_t)b * (NN + 1) + t) * (MM * DD));
        for (int i = tid; i < MM * DD / 4; i += 256) dst4[i] = mv4[i];
        __syncthreads();
        // read (pre-update)
        if (tid < DD) {
            float accv = 0.0f;
#pragma unroll
            for (int m = 0; m < MM; ++m) accv += wsh[m] * mv[m * DD + tid];
            Rd[(size_t)bt * DD + tid] = accv;
        }
        __syncthreads();
        // update
        for (int i = tid; i < MM * DD; i += 256) {
            int m = i >> 6;
            int d = i & 63;
            float wm = wsh[m];
            mv[i] = mv[i] * (1.0f - wm * esh[d]) + wm * ash[d];
        }
        __syncthreads();
    }
    // final state -> MvOut[b][NN]
    float4* dst4 = (float4*)(MvOut + ((size_t)b * (NN + 1) + NN) * (MM * DD));
    for (int i = tid; i < MM * DD / 4; i += 256) dst4[i] = mv4[i];
}

// ---------------------------------------------------------------------------
// 5) f = tanh(concat([read, K]) @ Wf + bf)  -- Kdim=128, N=64, f32 WMMA
// ---------------------------------------------------------------------------
__global__ void dkvmn_gemm_f(const float* __restrict__ Rd, const float* __restrict__ Kg,
                             const float* __restrict__ Wf, const float* __restrict__ bf,
                             float* __restrict__ F) {
    int wave = threadIdx.x >> 5;
    int lane = threadIdx.x & 31;
    int half = lane >> 4;
    int l    = lane & 15;
    int row0 = (blockIdx.x * 4 + wave) * 16;
    if (row0 >= TT) return;

    v8f acc[4] = {};
    const float* Rr = Rd + (size_t)(row0 + l) * DD;
    const float* Kr = Kg + (size_t)(row0 + l) * DD;

#pragma unroll
    for (int k0 = 0; k0 < 2 * DD; k0 += 4) {
        int k = k0 + 2 * half;      // k and k+1 never straddle the 64 boundary
        v2f a;
        if (k0 < DD) { a.x = Rr[k];      a.y = Rr[k + 1]; }
        else         { a.x = Kr[k - DD]; a.y = Kr[k - DD + 1]; }
#pragma unroll
        for (int nt = 0; nt < 4; ++nt) {
            int n = nt * 16 + l;
            v2f bv;
            bv.x = Wf[(size_t)(k + 0) * DD + n];
            bv.y = Wf[(size_t)(k + 1) * DD + n];
            acc[nt] = __builtin_amdgcn_wmma_f32_16x16x4_f32(
                false, a, false, bv, (short)0, acc[nt], false, false);
        }
    }

#pragma unroll
    for (int nt = 0; nt < 4; ++nt) {
        int n = nt * 16 + l;
        float bs = bf[n];
#pragma unroll
        for (int i = 0; i < 8; ++i) {
            int m = i + 8 * half;
            F[(size_t)(row0 + m) * DD + n] = tanhf(acc[nt][i] + bs);
        }
    }
}

// ---------------------------------------------------------------------------
// 6) p = sigmoid(f @ Wp + bp), one thread per token
// ---------------------------------------------------------------------------
__global__ void dkvmn_p(const float* __restrict__ F, const float* __restrict__ Wp,
                        const float* __restrict__ bp, float* __restrict__ p) {
    int t = blockIdx.x * 256 + threadIdx.x;
    if (t >= TT) return;
    float accv = bp[0];
    const float* fr = F + (size_t)t * DD;
#pragma unroll
    for (int d = 0; d < DD; ++d) accv += fr[d] * Wp[d];
    p[t] = 1.0f / (1.0f + __expf(-accv));
}

// ---------------------------------------------------------------------------
extern "C" void kernel_launch(void* const* d_in, const int* in_sizes, int n_in,
                              void* d_out, int out_size, void* d_ws, size_t ws_size,
                              hipStream_t stream) {
    const int*   q     = (const int*)d_in[0];
    const int*   r     = (const int*)d_in[1];
    // d_in[2] (at_s), d_in[3] (at_m): unused by the reference
    const float* k_emb = (const float*)d_in[4];
    const float* v_emb = (const float*)d_in[5];
    const float* Mk    = (const float*)d_in[6];
    const float* Mv0   = (const float*)d_in[7];
    const float* We    = (const float*)d_in[8];
    const float* be    = (const float*)d_in[9];
    const float* Wa    = (const float*)d_in[10];
    const float* ba    = (const float*)d_in[11];
    const float* Wf    = (const float*)d_in[12];
    const float* bf    = (const float*)d_in[13];
    const float* Wp    = (const float*)d_in[14];
    const float* bp    = (const float*)d_in[15];

    float* out    = (float*)d_out;
    float* p_out  = out;            // [128*200]
    float* Mv_out = out + TT;       // [128,201,50,64]

    // workspace layout (floats): K, V(->F), w, E, A, read = 370*TT ≈ 38 MB
    float* ws = (float*)d_ws;
    float* Kg = ws;
    float* Vg = Kg + (size_t)TT * DD;
    float* Wb = Vg + (size_t)TT * DD;
    float* Eg = Wb + (size_t)TT * MM;
    float* Ag = Eg + (size_t)TT * DD;
    float* Rd = Ag + (size_t)TT * DD;
    float* Fb = Vg;                 // reuse V buffer for f (V dead after E/A)

    int gemmBlocks = TT / (16 * 4); // 400 blocks, 4 waves each, 16 rows per wave

    dkvmn_gather<<<(TT * (DD / 4) + 255) / 256, 256, 0, stream>>>(q, r, k_emb, v_emb, Kg, Vg);
    // logits = K @ Mk^T  (N=50, transposed access into Mk[50,64], no bias, no act)
    dkvmn_gemm_act<MM, true, 0><<<gemmBlocks, 128, 0, stream>>>(Kg, Mk, nullptr, Wb);
    // E = sigmoid(V @ We + be), A = tanh(V @ Wa + ba)
    dkvmn_gemm_act<DD, false, 1><<<gemmBlocks, 128, 0, stream>>>(Vg, We, be, Eg);
    dkvmn_gemm_act<DD, false, 2><<<gemmBlocks, 128, 0, stream>>>(Vg, Wa, ba, Ag);
    dkvmn_softmax<<<(TT + 255) / 256, 256, 0, stream>>>(Wb);
    dkvmn_scan   <<<BB, 256, 0, stream>>>(Mv0, Wb, Eg, Ag, Rd, Mv_out);
    dkvmn_gemm_f <<<gemmBlocks, 128, 0, stream>>>(Rd, Kg, Wf, bf, Fb);
    dkvmn_p      <<<(TT + 255) / 256, 256, 0, stream>>>(Fb, Wp, bp, p_out);
}